// UncompressConvolution_76459007803908
// MI455X (gfx1250) — compile-verified
//
#include <hip/hip_runtime.h>

typedef float v2f __attribute__((ext_vector_type(2)));
typedef float v8f __attribute__((ext_vector_type(8)));

#define B_    4
#define H_    4
#define NC_   1024
#define DIN_  128
#define DM_   128
#define N_    4096        // NC_ * 4
#define TT_   32          // t-values (compressed positions) per workgroup (2 M-tiles)
#define LDSROWS (TT_ + 2) // need rows t0-1 .. t0+32

// Fused: zero-insert upsample + pad + unfold + Linear(+bias) + ReLU
//        + windowed residual (/2) - out + layer_query, exploiting the
//        2-of-8 tap sparsity of the transposed-conv structure.
// Each wave: one phase r, 4 N-tiles x 2 M-tiles (B-fragments reused across
// the two M-tiles -> halved W traffic, 8 WMMA per 6 loads).
__global__ __launch_bounds__(256)
void UncompressConvolution_kernel(const float* __restrict__ lq,   // [B,H,N,DM]
                                  const float* __restrict__ u,    // [B,H,NC,DIN]
                                  const float* __restrict__ W,    // [DM, 8*DIN]
                                  const float* __restrict__ bias, // [DM]
                                  float* __restrict__ out)        // [B,H,N,DM]
{
  __shared__ float lds_u[LDSROWS * DIN_];

  const int tid = threadIdx.x;
  const int bh  = blockIdx.x / (NC_ / TT_);
  const int tb  = blockIdx.x % (NC_ / TT_);
  const int t0  = tb * TT_;

  // ---- stage u rows [t0-1, t0+32] into LDS, OOB rows -> 0 ----
  const float* ubh = u + (size_t)bh * NC_ * DIN_;
  for (int i = tid; i < LDSROWS * DIN_; i += 256) {
    const int row = i >> 7;        // /128
    const int col = i & 127;
    const int nc  = t0 - 1 + row;
    float v = 0.0f;
    if (nc >= 0 && nc < NC_) v = ubh[(size_t)nc * DIN_ + col];
    lds_u[i] = v;
  }
  __syncthreads();

  const int wave = tid >> 5;      // wave32
  const int lane = tid & 31;
  const int r    = wave & 3;      // phase within stride
  const int ng   = wave >> 2;     // which half of the 8 N-tiles
  const int tl   = lane & 15;     // spatial index within 16x16 tile
  const int kh   = lane >> 4;     // K-half select for A/B fragments

  // filter-tap columns and row offsets per phase r
  const int fa_tab[4] = {0, 3, 2, 1};
  const int fa = fa_tab[r];
  const int fb = fa + 4;
  const int roff_lo = (r == 0) ? 0 : 1;  // LDS row offset for tap a
  const int roff_hi = roff_lo + 1;       // for tap b

  v8f a00 = {}, a01 = {}, a02 = {}, a03 = {};  // M-tile 0 (t0..t0+15)
  v8f a10 = {}, a11 = {}, a12 = {}, a13 = {};  // M-tile 1 (t0+16..t0+31)

  const int n0 = ng * 4;  // first of this wave's 4 N-tiles
  const float* wr0 = W + (size_t)((n0 + 0) * 16 + tl) * (8 * DIN_);
  const float* wr1 = W + (size_t)((n0 + 1) * 16 + tl) * (8 * DIN_);
  const float* wr2 = W + (size_t)((n0 + 2) * 16 + tl) * (8 * DIN_);
  const float* wr3 = W + (size_t)((n0 + 3) * 16 + tl) * (8 * DIN_);

  // K = 256: first 128 from tap (fa, nc_a), second 128 from tap (fb, nc_b)
  #pragma unroll
  for (int half = 0; half < 2; ++half) {
    const int f    = half ? fb : fa;
    const int roff = roff_lo + half;
    const float* arow0 = &lds_u[(tl + roff) * DIN_];        // A rows, M-tile 0
    const float* arow1 = &lds_u[(tl + 16 + roff) * DIN_];   // A rows, M-tile 1
    const float* wcol0 = wr0 + f * DIN_;
    const float* wcol1 = wr1 + f * DIN_;
    const float* wcol2 = wr2 + f * DIN_;
    const float* wcol3 = wr3 + f * DIN_;
    #pragma unroll 2
    for (int kk = 0; kk < 32; ++kk) {
      const int k = kk * 4 + 2 * kh;  // per-lane K base (v=0); v=1 is k+1
      const v2f av0 = *(const v2f*)(arow0 + k);  // ds_load_b64
      const v2f av1 = *(const v2f*)(arow1 + k);
      const v2f b0  = *(const v2f*)(wcol0 + k);  // global_load_b64, L2-resident W
      const v2f b1  = *(const v2f*)(wcol1 + k);
      const v2f b2  = *(const v2f*)(wcol2 + k);
      const v2f b3  = *(const v2f*)(wcol3 + k);
      a00 = __builtin_amdgcn_wmma_f32_16x16x4_f32(false, av0, false, b0, (short)0, a00, false, false);
      a10 = __builtin_amdgcn_wmma_f32_16x16x4_f32(false, av1, false, b0, (short)0, a10, false, false);
      a01 = __builtin_amdgcn_wmma_f32_16x16x4_f32(false, av0, false, b1, (short)0, a01, false, false);
      a11 = __builtin_amdgcn_wmma_f32_16x16x4_f32(false, av1, false, b1, (short)0, a11, false, false);
      a02 = __builtin_amdgcn_wmma_f32_16x16x4_f32(false, av0, false, b2, (short)0, a02, false, false);
      a12 = __builtin_amdgcn_wmma_f32_16x16x4_f32(false, av1, false, b2, (short)0, a12, false, false);
      a03 = __builtin_amdgcn_wmma_f32_16x16x4_f32(false, av0, false, b3, (short)0, a03, false, false);
      a13 = __builtin_amdgcn_wmma_f32_16x16x4_f32(false, av1, false, b3, (short)0, a13, false, false);
    }
  }

  // ---- epilogue: bias + relu, windowed residual /2, + layer_query ----
  const float* lqbh = lq  + (size_t)bh * N_ * DM_;
  float*       obh  = out + (size_t)bh * N_ * DM_;

  auto epi = [&](v8f acc, int mt, int j) {
    const int   c  = (n0 + j) * 16 + tl;   // output channel (N index of tile)
    const float bv = bias[c];
    #pragma unroll
    for (int v = 0; v < 8; ++v) {
      const int   tlo = mt * 16 + v + 8 * kh; // C layout: M = v (lanes 0-15) / v+8 (16-31)
      const int   p   = 4 * (t0 + tlo) + r;
      const float uo  = fmaxf(acc[v] + bv, 0.0f);
      const float ua  = lds_u[(tlo + roff_lo) * DIN_ + c];
      const float ub  = lds_u[(tlo + roff_hi) * DIN_ + c];
      const float us  = 0.5f * (ua + ub);
      const size_t idx = (size_t)p * DM_ + c;
      obh[idx] = us - uo + lqbh[idx];
    }
  };
  epi(a00, 0, 0); epi(a01, 0, 1); epi(a02, 0, 2); epi(a03, 0, 3);
  epi(a10, 1, 0); epi(a11, 1, 1); epi(a12, 1, 2); epi(a13, 1, 3);
}

extern "C" void kernel_launch(void* const* d_in, const int* in_sizes, int n_in,
                              void* d_out, int out_size, void* d_ws, size_t ws_size,
                              hipStream_t stream) {
  (void)in_sizes; (void)n_in; (void)out_size; (void)d_ws; (void)ws_size;
  const float* lq   = (const float*)d_in[0];  // layer_query  [B,H,N,DM]
  const float* u    = (const float*)d_in[1];  // uncompress_in[B,H,NC,DIN]
  const float* W    = (const float*)d_in[2];  // [DM, DIN*FS]
  const float* bias = (const float*)d_in[3];  // [DM]
  // d_in[4] = i (unused)
  float* out = (float*)d_out;

  const dim3 grid(B_ * H_ * (NC_ / TT_));  // 512 workgroups
  const dim3 block(256);                   // 8 wave32 waves
  hipLaunchKernelGGL(UncompressConvolution_kernel, grid, block, 0, stream,
                     lq, u, W, bias, out);
}